// vGPSSM_40905268527436
// MI455X (gfx1250) — compile-verified
//
#include <hip/hip_runtime.h>
#include <hip/hip_bf16.h>

// ---------------------------------------------------------------------------
// vGPSSM ELBO on MI455X (gfx1250).  Sequential scan -> host loop enqueues two
// fused kernels per step; all carried state lives in d_ws (L2-resident).
// 16x16 f32 GEMMs -> v_wmma_f32_16x16x4_f32 chains (wave32, one wave/block).
// ---------------------------------------------------------------------------

#ifndef __has_builtin
#define __has_builtin(x) 0
#endif
#if __has_builtin(__builtin_amdgcn_wmma_f32_16x16x4_f32)
#define HAVE_WMMA_F32 1
#else
#define HAVE_WMMA_F32 0
#endif

typedef __attribute__((ext_vector_type(2))) float v2f;
typedef __attribute__((ext_vector_type(8))) float v8f;

constexpr int NB = 256;   // batch
constexpr int NT = 256;   // time steps
constexpr int ND = 16;    // state dim
constexpr int NDY = 8;    // obs dim
constexpr int NM = 64;    // inducing points
__device__ constexpr float LOG2PI_F = 1.8378770664093453f;
__device__ constexpr float JITTER_Q = 1e-3f;
__device__ constexpr float JITTER_K = 1e-4f;

// ---- workspace layout (floats) --------------------------------------------
constexpr size_t ACC_OFF = 0;                       // 16 accumulators
constexpr size_t X_OFF   = 16;                      // 2 * B*D (double buffer)
constexpr size_t M_OFF   = X_OFF + 2 * NB * ND;     // B*D
constexpr size_t SIG_OFF = M_OFF + NB * ND;         // B*D*D
constexpr size_t KINV_OFF= SIG_OFF + NB * ND * ND;  // D*M*M
constexpr size_t WS_FLOATS = KINV_OFF + ND * NM * NM;
// acc indices: 0=qm0KL sum, 1=logdetL sum, 2=KL_u, 3=data_fit, 4=gp_dyn

// ---- WMMA fragment helpers (ISA 7.12.2 layouts, wave32) --------------------
__device__ inline void store_c16(float* C, int ldc, int lane, v8f c) {
  int col = lane & 15;
  int rbase = (lane >> 4) << 3;
#pragma unroll
  for (int g = 0; g < 8; ++g) C[(rbase + g) * ldc + col] = c[g];
}

// Accumulate one 16x16x4 chunk: C += A(:,kc*4..+3) * B(kc*4..+3,:)
// trans flags mean the *storage* is transposed relative to the logical matrix.
__device__ inline void wmma_acc4(v8f& c, const float* A, int lda, bool tA,
                                 const float* Bm, int ldb, bool tB,
                                 int kc, int lane) {
#if HAVE_WMMA_F32
  int row = lane & 15;
  int col = lane & 15;
  int k0 = kc * 4 + ((lane >> 4) << 1);
  v2f a, b;
  if (!tA) { a.x = A[row * lda + k0];     a.y = A[row * lda + k0 + 1]; }
  else     { a.x = A[k0 * lda + row];     a.y = A[(k0 + 1) * lda + row]; }
  if (!tB) { b.x = Bm[k0 * ldb + col];    b.y = Bm[(k0 + 1) * ldb + col]; }
  else     { b.x = Bm[col * ldb + k0];    b.y = Bm[col * ldb + k0 + 1]; }
  c = __builtin_amdgcn_wmma_f32_16x16x4_f32(false, a, false, b,
                                            (short)0, c, false, false);
#else
  int col = lane & 15;
  int rbase = (lane >> 4) << 3;
#pragma unroll
  for (int g = 0; g < 8; ++g) {
    float s = 0.f;
#pragma unroll
    for (int kk = 0; kk < 4; ++kk) {
      int k = kc * 4 + kk;
      float av = tA ? A[k * lda + (rbase + g)] : A[(rbase + g) * lda + k];
      float bv = tB ? Bm[col * ldb + k] : Bm[k * ldb + col];
      s += av * bv;
    }
    c[g] += s;
  }
#endif
}

__device__ inline float wave_reduce(float v) {
#pragma unroll
  for (int off = 16; off; off >>= 1) v += __shfl_xor(v, off, 32);
  return v;
}

// ---------------------------------------------------------------------------
__global__ void zero_acc_kernel(float* ws) {
  if (threadIdx.x < 16) ws[ACC_OFF + threadIdx.x] = 0.f;
}

// qm0 KL + entropy logdet reductions (one shot)
__global__ void scalar_terms_kernel(const float* __restrict__ qm,
                                    const float* __restrict__ qv,
                                    const float* __restrict__ Lt_all,
                                    float* ws) {
  int idx = blockIdx.x * blockDim.x + threadIdx.x;
  float ld = 0.f, q = 0.f;
  if (idx < NB * NT) {
    const float* L = Lt_all + (size_t)idx * (ND * ND);
#pragma unroll
    for (int d = 0; d < ND; ++d) ld += __logf(L[d * (ND + 1)]);
  }
  if (idx < NB) {
#pragma unroll
    for (int d = 0; d < ND; ++d) {
      float v = qv[idx * ND + d], m = qm[idx * ND + d];
      q += 0.5f * (v + m * m - 1.f - __logf(v));
    }
  }
  ld = wave_reduce(ld);
  q = wave_reduce(q);
  if ((threadIdx.x & 31) == 0) {
    atomicAdd(&ws[ACC_OFF + 1], ld);
    atomicAdd(&ws[ACC_OFF + 0], q);
  }
}

// x0 = qm + sqrt(v)*eps0 ; m0 = qm ; Sig0 = diag(v)
__global__ void init_state_kernel(const float* __restrict__ qm,
                                  const float* __restrict__ qv,
                                  const float* __restrict__ eps0,
                                  float* ws) {
  int idx = blockIdx.x * blockDim.x + threadIdx.x;
  if (idx < NB * ND * ND) {
    int b = idx >> 8, rc = idx & 255;
    int r = rc >> 4, c = rc & 15;
    ws[SIG_OFF + idx] = (r == c) ? qv[b * ND + r] : 0.f;
  }
  if (idx < NB * ND) {
    float m = qm[idx], v = qv[idx];
    ws[M_OFF + idx] = m;
    ws[X_OFF + idx] = m + sqrtf(v) * eps0[idx];  // x buffer parity 0
  }
}

// Per-task: Kzz, Cholesky, Lzz^-1, Kzz^-1 -> ws, and KL[q(u)||p(u)] pieces.
__global__ __launch_bounds__(64) void task_precompute_kernel(
    const float* __restrict__ Z, const float* __restrict__ m_u,
    const float* __restrict__ Lu, const float* __restrict__ log_ls,
    const float* __restrict__ log_os, float* ws) {
  __shared__ float sK[NM * NM];
  __shared__ float sI[NM * NM];
  __shared__ float red[NM];
  const int task = blockIdx.x, tid = threadIdx.x;
  const float ls2 = __expf(2.f * log_ls[task]);
  const float os_ = __expf(log_os[task]);
  const float* Zt = Z + (size_t)task * NM * ND;
  const float* Lut = Lu + (size_t)task * NM * NM;
  // Kzz + jitter
  for (int e = tid; e < NM * NM; e += 64) {
    int i = e >> 6, j = e & 63;
    float d2 = 0.f;
#pragma unroll
    for (int d = 0; d < ND; ++d) {
      float df = Zt[i * ND + d] - Zt[j * ND + d];
      d2 += df * df;
    }
    float v = os_ * __expf(-0.5f * d2 / ls2);
    if (i == j) v += JITTER_K;
    sK[e] = v;
  }
  __syncthreads();
  // in-place lower Cholesky
  for (int j = 0; j < NM; ++j) {
    if (tid == 0) sK[j * NM + j] = sqrtf(sK[j * NM + j]);
    __syncthreads();
    if (tid > j) sK[tid * NM + j] /= sK[j * NM + j];
    __syncthreads();
    if (tid > j) {
      float lij = sK[tid * NM + j];
      for (int k = j + 1; k <= tid; ++k) sK[tid * NM + k] -= lij * sK[k * NM + j];
    }
    __syncthreads();
  }
  // Lzz^-1: thread = column
  {
    int j = tid;
    for (int i = 0; i < j; ++i) sI[i * NM + j] = 0.f;
    sI[j * NM + j] = 1.f / sK[j * NM + j];
    for (int i = j + 1; i < NM; ++i) {
      float s = 0.f;
      for (int k = j; k < i; ++k) s += sK[i * NM + k] * sI[k * NM + j];
      sI[i * NM + j] = -s / sK[i * NM + i];
    }
  }
  __syncthreads();
  // Kzz^-1 = Linv^T Linv -> global ws
  float* Kinv = ws + KINV_OFF + (size_t)task * NM * NM;
  for (int e = tid; e < NM * NM; e += 64) {
    int i = e >> 6, j = e & 63;
    float s = 0.f;
    for (int k = 0; k < NM; ++k) s += sI[k * NM + i] * sI[k * NM + j];
    Kinv[e] = s;
  }
  // KL pieces: ||Linv Lu||_F^2, ||Linv m_u||^2, logdets
  float part = 0.f;
  for (int e = tid; e < NM * NM; e += 64) {
    int i = e >> 6, j = e & 63;
    float s = 0.f;
    for (int k = j; k <= i; ++k) s += sI[i * NM + k] * Lut[k * NM + j];
    part += s * s;
  }
  {
    float s = 0.f;
    const float* mut = m_u + task * NM;
    for (int k = 0; k <= tid; ++k) s += sI[tid * NM + k] * mut[k];
    part += s * s;
    part += 2.f * __logf(sK[tid * NM + tid]) - 2.f * __logf(Lut[tid * NM + tid]);
  }
  red[tid] = part;
  __syncthreads();
  if (tid == 0) {
    float s = 0.f;
    for (int k = 0; k < NM; ++k) s += red[k];
    atomicAdd(&ws[ACC_OFF + 2], 0.5f * (s - (float)NM));
  }
}

// ---- per-step filter: one wave per batch ----------------------------------
__global__ __launch_bounds__(32) void filter_step_kernel(
    const float* __restrict__ At_all, const float* __restrict__ Lt_all,
    const float* __restrict__ obs, const float* __restrict__ eps,
    const float* __restrict__ H, const float* __restrict__ s2e,
    float* ws, float* x_out, int t) {
  __shared__ float sA[256], sL[256], sS[256], sT[256], sN[256];
  __shared__ float sm[16], smt[16];
  const int b = blockIdx.x, lane = threadIdx.x;
  const float* Ab = At_all + ((size_t)b * NT + t) * 256;
  const float* Lb = Lt_all + ((size_t)b * NT + t) * 256;
  float* Sig = ws + SIG_OFF + (size_t)b * 256;
  float* mst = ws + M_OFF + b * ND;
  for (int i = lane; i < 256; i += 32) {
    sA[i] = Ab[i]; sL[i] = Lb[i]; sS[i] = Sig[i];
  }
  if (lane < 16) sm[lane] = mst[lane];
  if (t + 1 < NT) {  // warm next step's tiles (global_prefetch_b8)
    __builtin_prefetch(Ab + 256 + lane * 8, 0, 0);
    __builtin_prefetch(Lb + 256 + lane * 8, 0, 0);
  }
  __syncthreads();
  if (lane < 16) {
    float s = 0.f;
#pragma unroll
    for (int d = 0; d < 16; ++d) s += sA[lane * 16 + d] * sm[d];
    smt[lane] = s;
  }
  __syncthreads();
  // Tmp = A * Sig_prev
  v8f cT = {};
#pragma unroll
  for (int kc = 0; kc < 4; ++kc) wmma_acc4(cT, sA, 16, false, sS, 16, false, kc, lane);
  store_c16(sT, 16, lane, cT);
  __syncthreads();
  // Sig = Tmp * A^T + L * L^T
  v8f cS = {};
#pragma unroll
  for (int kc = 0; kc < 4; ++kc) wmma_acc4(cS, sT, 16, false, sA, 16, true, kc, lane);
#pragma unroll
  for (int kc = 0; kc < 4; ++kc) wmma_acc4(cS, sL, 16, false, sL, 16, true, kc, lane);
  store_c16(sN, 16, lane, cS);
  __syncthreads();
  if (lane < 16) sN[lane * 17] += JITTER_Q;
  __syncthreads();
  for (int i = lane; i < 256; i += 32) Sig[i] = sN[i];
  // data fit: logp - 0.5 * tr(H Sig H^T / s2e), generic H
  float dpart = 0.f;
  if (lane < NDY) {
    float cm = 0.f, hsh = 0.f;
#pragma unroll
    for (int d = 0; d < 16; ++d) {
      float hd = H[lane * 16 + d];
      cm += hd * smt[d];
      float tr = 0.f;
#pragma unroll
      for (int e = 0; e < 16; ++e) tr += sN[d * 16 + e] * H[lane * 16 + e];
      hsh += hd * tr;
    }
    float y = obs[((size_t)b * NT + t) * NDY + lane];
    float r = y - cm;
    float s2 = s2e[lane];
    dpart = -0.5f * (r * r / s2 + __logf(6.283185307179586f * s2))
            - 0.5f * hsh / s2;
  }
  dpart = wave_reduce(dpart);
  if (lane == 0) atomicAdd(&ws[ACC_OFF + 3], dpart / (float)NB);
  // 16x16 Cholesky of sN in place (lower)
  for (int j = 0; j < 16; ++j) {
    if (lane == 0) sN[j * 17] = sqrtf(sN[j * 17]);
    __syncthreads();
    if (lane > j && lane < 16) sN[lane * 16 + j] /= sN[j * 17];
    __syncthreads();
    if (lane > j && lane < 16) {
      float lij = sN[lane * 16 + j];
      for (int k = j + 1; k <= lane; ++k) sN[lane * 16 + k] -= lij * sN[k * 16 + j];
    }
    __syncthreads();
  }
  // x_t = mt + L * eps_t ; commit carried state
  if (lane < 16) {
    const float* ep = eps + ((size_t)t * NB + b) * ND;
    float s = smt[lane];
    for (int d = 0; d <= lane; ++d) s += sN[lane * 16 + d] * ep[d];
    x_out[b * ND + lane] = s;
    mst[lane] = smt[lane];
  }
}

// ---- per-step SVGP term: one wave per (task, 16-batch tile) ---------------
__global__ __launch_bounds__(32) void gp_step_kernel(
    const float* __restrict__ Z, const float* __restrict__ m_u,
    const float* __restrict__ Lu, const float* __restrict__ log_ls,
    const float* __restrict__ log_os, const float* __restrict__ s2_proc,
    const float* __restrict__ x_prev, const float* __restrict__ x_curr,
    float* ws) {
  __shared__ float sX[16 * 16];
  __shared__ float sKzx[NM * 16];
  __shared__ float sW[NM * 16];
  const int lane = threadIdx.x;
  const int task = blockIdx.x >> 4;
  const int b0 = (blockIdx.x & 15) * 16;
  const float ls2 = __expf(2.f * log_ls[task]);
  const float os_ = __expf(log_os[task]);
  const float s2p = s2_proc[task];
  for (int i = lane; i < 256; i += 32) {
    int bl = i >> 4, d = i & 15;
    sX[i] = x_prev[(b0 + bl) * ND + d];
  }
  __syncthreads();
  const float* Zt = Z + (size_t)task * NM * ND;
  for (int e = lane; e < NM * 16; e += 32) {
    int m = e >> 4, bl = e & 15;
    float d2 = 0.f;
#pragma unroll
    for (int d = 0; d < ND; ++d) {
      float df = sX[bl * 16 + d] - Zt[m * ND + d];
      d2 += df * df;
    }
    sKzx[e] = os_ * __expf(-0.5f * d2 / ls2);
  }
  __syncthreads();
  // w = Kzz^-1 * Kzx   (64x64 * 64x16, WMMA chains)
  const float* Kt = ws + KINV_OFF + (size_t)task * NM * NM;
  for (int r = 0; r < 4; ++r) {
    v8f c = {};
    for (int kc = 0; kc < 16; ++kc)
      wmma_acc4(c, Kt + r * 16 * NM, NM, false, sKzx, 16, false, kc, lane);
    store_c16(sW + r * 16 * 16, 16, lane, c);
  }
  __syncthreads();
  // u = Lu^T * w ; accumulate sum_n u[n,b]^2 straight from fragments
  const float* Lut = Lu + (size_t)task * NM * NM;
  float uu = 0.f;
  for (int r = 0; r < 4; ++r) {
    v8f c = {};
    for (int kc = 0; kc < 16; ++kc)
      wmma_acc4(c, Lut + r * 16, NM, true, sW, 16, false, kc, lane);
#pragma unroll
    for (int g = 0; g < 8; ++g) uu += c[g] * c[g];
  }
  uu += __shfl_xor(uu, 16, 32);  // lanes l and l+16 hold the same column
  float part = 0.f;
  if (lane < 16) {
    int bl = lane;
    float kw = 0.f, mu = 0.f;
    const float* mut = m_u + task * NM;
    for (int m = 0; m < NM; ++m) {
      float wv = sW[m * 16 + bl];
      kw += sKzx[m * 16 + bl] * wv;
      mu += mut[m] * wv;
    }
    float var = os_ - kw + uu;
    var = var > 1e-8f ? var : 1e-8f;
    float xt = x_curr[(b0 + bl) * ND + task];
    float d = xt - mu;
    part = -0.5f * (LOG2PI_F + __logf(s2p) + (d * d + var) / s2p);
  }
  part = wave_reduce(part);
  if (lane == 0)
    atomicAdd(&ws[ACC_OFF + 4], part / (float)(ND * NB * NT));
}

__global__ void finalize_kernel(const float* ws, float* out) {
  float qm0 = ws[ACC_OFF + 0] / (float)NB;
  float Hx = (0.5f * (float)(NT * ND) * LOG2PI_F + ws[ACC_OFF + 1] / (float)NB
              + 0.5f * (float)(NT * ND)) / (float)NT;
  float kld = ws[ACC_OFF + 2] / (float)(NT * NB);
  out[0] = -qm0 + Hx - kld + ws[ACC_OFF + 3] + ws[ACC_OFF + 4];
}

// ---------------------------------------------------------------------------
extern "C" void kernel_launch(void* const* d_in, const int* in_sizes, int n_in,
                              void* d_out, int out_size, void* d_ws, size_t ws_size,
                              hipStream_t stream) {
  if (ws_size < WS_FLOATS * sizeof(float)) return;
  const float* obs     = (const float*)d_in[0];
  const float* At_all  = (const float*)d_in[1];
  const float* Lt_all  = (const float*)d_in[2];
  const float* qx0_m   = (const float*)d_in[3];
  const float* qx0_v   = (const float*)d_in[4];
  const float* eps0    = (const float*)d_in[5];
  const float* eps     = (const float*)d_in[6];
  const float* H       = (const float*)d_in[7];
  const float* Z       = (const float*)d_in[8];
  const float* m_u     = (const float*)d_in[9];
  const float* Lu      = (const float*)d_in[10];
  const float* log_ls  = (const float*)d_in[11];
  const float* log_os  = (const float*)d_in[12];
  const float* s2_proc = (const float*)d_in[13];
  const float* s2_emis = (const float*)d_in[14];
  float* ws  = (float*)d_ws;
  float* out = (float*)d_out;

  zero_acc_kernel<<<1, 32, 0, stream>>>(ws);
  init_state_kernel<<<(NB * ND * ND + 255) / 256, 256, 0, stream>>>(
      qx0_m, qx0_v, eps0, ws);
  scalar_terms_kernel<<<(NB * NT + 255) / 256, 256, 0, stream>>>(
      qx0_m, qx0_v, Lt_all, ws);
  task_precompute_kernel<<<ND, 64, 0, stream>>>(Z, m_u, Lu, log_ls, log_os, ws);

  for (int t = 0; t < NT; ++t) {
    float* x_prev = ws + X_OFF + (size_t)(t & 1) * NB * ND;
    float* x_next = ws + X_OFF + (size_t)((t + 1) & 1) * NB * ND;
    filter_step_kernel<<<NB, 32, 0, stream>>>(
        At_all, Lt_all, obs, eps, H, s2_emis, ws, x_next, t);
    gp_step_kernel<<<ND * (NB / 16), 32, 0, stream>>>(
        Z, m_u, Lu, log_ls, log_os, s2_proc, x_prev, x_next, ws);
  }
  finalize_kernel<<<1, 1, 0, stream>>>(ws, out);
}